// WithinSubjectTripletLoss_18751827214370
// MI455X (gfx1250) — compile-verified
//
#include <hip/hip_runtime.h>

#define B_N    8192
#define D_DIM  128
#define MARGIN 0.8f
#define EPSV   1e-6f

typedef float v2f __attribute__((ext_vector_type(2)));
typedef float v8f __attribute__((ext_vector_type(8)));

// ---------------------------------------------------------------------------
// Kernel 0: per-row squared norm + packed (sbj<<8)|label
// ---------------------------------------------------------------------------
__global__ __launch_bounds__(256)
void precomp_kernel(const float* __restrict__ emb,
                    const int*   __restrict__ labels,
                    const int*   __restrict__ sbj,
                    float* __restrict__ sq,
                    int*   __restrict__ packed) {
  int i = blockIdx.x * blockDim.x + threadIdx.x;
  if (i >= B_N) return;
  const float4* row = (const float4*)(emb + i * D_DIM);
  float s = 0.f;
#pragma unroll
  for (int k = 0; k < D_DIM / 4; ++k) {
    float4 v = row[k];
    s += v.x * v.x + v.y * v.y + v.z * v.z + v.w * v.w;
  }
  sq[i] = s;
  packed[i] = (sbj[i] << 8) | labels[i];
}

// ---------------------------------------------------------------------------
// Kernel 1: fused Gram-matrix + hardest-positive/negative mining.
//
// 2-D decomposition for occupancy: each 256-thread WG owns ONE 16-row tile;
// its 8 waves split the 8192-column sweep 8 ways (1024 columns each), using
// V_WMMA_F32_16X16X4_F32 (K=128 -> 32 wmma per 16x16 tile, 4 independent
// accumulator chains). 512 WGs / 4096 waves fill the chip; emb (4 MB) stays
// L2-resident so the column re-streaming is cheap.
//
// Mining score: d2[i,j] - sq[i] = sq[j] - 2*dot(i,j)  (row-constant dropped).
// Intra-wave butterfly reduce (16-lane halves per the C/D VGPR layout), then
// a deterministic LDS merge of the 8 per-wave partials per row.
// ---------------------------------------------------------------------------
__global__ __launch_bounds__(256)
void mine_kernel(const float* __restrict__ emb,
                 const float* __restrict__ sq,
                 const int*   __restrict__ packed,
                 int* __restrict__ p_idx,
                 int* __restrict__ n_idx) {
  const int lane    = threadIdx.x & 31;
  const int wave    = threadIdx.x >> 5;
  const int rowBase = blockIdx.x * 16;                // 16 rows per WG
  const int halfsel = lane >> 4;                      // 0: lanes 0-15, 1: 16-31
  const int l15     = lane & 15;
  const int koff    = halfsel * 2;                    // K sub-offset (A/B ISA layout)

  __shared__ float sPV[8][16];
  __shared__ int   sPI[8][16];
  __shared__ float sNV[8][16];
  __shared__ int   sNI[8][16];

  // A tile: row = rowBase + l15 ; chunk k holds K = 4k+koff, 4k+koff+1
  v2f a[32];
  {
    const float* rp = emb + (size_t)(rowBase + l15) * D_DIM + koff;
#pragma unroll
    for (int k = 0; k < 32; ++k)
      a[k] = *(const v2f*)(rp + 4 * k);
  }

  // Row-side packed ids for this lane's 8 rows: r(v) = rowBase + v + 8*halfsel
  int pr[8];
#pragma unroll
  for (int v = 0; v < 8; ++v)
    pr[v] = packed[rowBase + v + 8 * halfsel];

  float bestP[8], bestN[8];
  int   bpi[8],  bni[8];
#pragma unroll
  for (int v = 0; v < 8; ++v) {
    bestP[v] = -__builtin_inff();
    bestN[v] =  __builtin_inff();
    bpi[v] = 0; bni[v] = 0;
  }

  // This wave's column range: 1024 columns = 64 tiles of 16.
  const int colBegin = wave * (B_N / 8);
  const int colEnd   = colBegin + (B_N / 8);

  for (int c = colBegin; c < colEnd; c += 16) {
    const int   j   = c + l15;                 // this lane's column
    const int   pj  = packed[j];
    const float sqj = sq[j];
    const float* cp = emb + (size_t)j * D_DIM + koff;

    v8f acc0 = {}; v8f acc1 = {}; v8f acc2 = {}; v8f acc3 = {};
#pragma unroll
    for (int k = 0; k < 32; k += 4) {
      v2f b0 = *(const v2f*)(cp + 4 * (k + 0));
      v2f b1 = *(const v2f*)(cp + 4 * (k + 1));
      v2f b2 = *(const v2f*)(cp + 4 * (k + 2));
      v2f b3 = *(const v2f*)(cp + 4 * (k + 3));
      acc0 = __builtin_amdgcn_wmma_f32_16x16x4_f32(false, a[k + 0], false, b0,
                                                   (short)0, acc0, false, false);
      acc1 = __builtin_amdgcn_wmma_f32_16x16x4_f32(false, a[k + 1], false, b1,
                                                   (short)0, acc1, false, false);
      acc2 = __builtin_amdgcn_wmma_f32_16x16x4_f32(false, a[k + 2], false, b2,
                                                   (short)0, acc2, false, false);
      acc3 = __builtin_amdgcn_wmma_f32_16x16x4_f32(false, a[k + 3], false, b3,
                                                   (short)0, acc3, false, false);
    }
    v8f dot = (acc0 + acc1) + (acc2 + acc3);

    // Branchless mining update (EXEC stays all-1s around the WMMA loop).
#pragma unroll
    for (int v = 0; v < 8; ++v) {
      const int   r     = rowBase + v + 8 * halfsel;
      const float val   = fmaf(-2.0f, dot[v], sqj);
      const int   x     = pj ^ pr[v];
      const bool  sameS = (x >> 8) == 0;
      const bool  sameL = (x == 0);
      const bool  isPos = sameS &&  sameL && (j != r);
      const bool  isNeg = sameS && !sameL;
      const bool  uP    = isPos && (val > bestP[v]);
      const bool  uN    = isNeg && (val < bestN[v]);
      bestP[v] = uP ? val : bestP[v];
      bpi[v]   = uP ? j   : bpi[v];
      bestN[v] = uN ? val : bestN[v];
      bni[v]   = uN ? j   : bni[v];
    }
  }

  // Butterfly reduce across the 16 lanes of each half (C/D layout: VGPR v ->
  // row v (lanes 0-15) or row v+8 (lanes 16-31), lane -> column).
#pragma unroll
  for (int v = 0; v < 8; ++v) {
#pragma unroll
    for (int m = 1; m < 16; m <<= 1) {
      float oPV = __shfl_xor(bestP[v], m, 16);
      int   oPI = __shfl_xor(bpi[v],   m, 16);
      float oNV = __shfl_xor(bestN[v], m, 16);
      int   oNI = __shfl_xor(bni[v],   m, 16);
      bool uP = oPV > bestP[v];
      bool uN = oNV < bestN[v];
      bestP[v] = uP ? oPV : bestP[v];
      bpi[v]   = uP ? oPI : bpi[v];
      bestN[v] = uN ? oNV : bestN[v];
      bni[v]   = uN ? oNI : bni[v];
    }
  }

  // Dump this wave's per-row partials to LDS (lane l15 == v owns row
  // v + 8*halfsel within the tile).
#pragma unroll
  for (int v = 0; v < 8; ++v) {
    if (l15 == v) {
      const int rr = v + 8 * halfsel;
      sPV[wave][rr] = bestP[v];
      sPI[wave][rr] = bpi[v];
      sNV[wave][rr] = bestN[v];
      sNI[wave][rr] = bni[v];
    }
  }
  __syncthreads();

  // Deterministic merge of the 8 wave-partials per row (fixed wave order).
  if (threadIdx.x < 16) {
    const int rr = threadIdx.x;
    const int r  = rowBase + rr;
    float bP = -__builtin_inff(); int bPI = 0;
    float bN =  __builtin_inff(); int bNI = 0;
#pragma unroll
    for (int w = 0; w < 8; ++w) {
      float pv = sPV[w][rr]; int pi = sPI[w][rr];
      float nv = sNV[w][rr]; int ni = sNI[w][rr];
      bool uP = pv > bP;
      bool uN = nv < bN;
      bP = uP ? pv : bP; bPI = uP ? pi : bPI;
      bN = uN ? nv : bN; bNI = uN ? ni : bNI;
    }
    const bool valid = (bP > -__builtin_inff()) && (bN < __builtin_inff());
    p_idx[r] = valid ? bPI : -1;
    n_idx[r] = valid ? bNI : -1;
  }
}

// ---------------------------------------------------------------------------
// Kernel 2: exact fp32 triplet hinge per row (one wave per row).
// dap = ||e_i - e_p + eps||, dan = ||e_i - e_n + eps|| (eps added elementwise,
// matching torch pairwise_distance / the reference).
// ---------------------------------------------------------------------------
__global__ __launch_bounds__(256)
void triplet_kernel(const float* __restrict__ emb,
                    const int*   __restrict__ p_idx,
                    const int*   __restrict__ n_idx,
                    float* __restrict__ per,
                    float* __restrict__ cnt) {
  const int lane = threadIdx.x & 31;
  const int wave = threadIdx.x >> 5;
  const int i    = blockIdx.x * 8 + wave;
  const int p    = p_idx[i];
  const int n    = n_idx[i];
  if (p < 0) {
    if (lane == 0) { per[i] = 0.f; cnt[i] = 0.f; }
    return;
  }
  const float4 e  = ((const float4*)(emb + (size_t)i * D_DIM))[lane];
  const float4 ep = ((const float4*)(emb + (size_t)p * D_DIM))[lane];
  const float4 en = ((const float4*)(emb + (size_t)n * D_DIM))[lane];
  float sp, sn;
  {
    float dx = e.x - ep.x + EPSV, dy = e.y - ep.y + EPSV;
    float dz = e.z - ep.z + EPSV, dw = e.w - ep.w + EPSV;
    sp = dx * dx + dy * dy + dz * dz + dw * dw;
  }
  {
    float dx = e.x - en.x + EPSV, dy = e.y - en.y + EPSV;
    float dz = e.z - en.z + EPSV, dw = e.w - en.w + EPSV;
    sn = dx * dx + dy * dy + dz * dz + dw * dw;
  }
#pragma unroll
  for (int m = 1; m < 32; m <<= 1) {
    sp += __shfl_xor(sp, m, 32);
    sn += __shfl_xor(sn, m, 32);
  }
  if (lane == 0) {
    float v = sqrtf(sp) - sqrtf(sn) + MARGIN;
    per[i] = v > 0.f ? v : 0.f;
    cnt[i] = 1.f;
  }
}

// ---------------------------------------------------------------------------
// Kernel 3: deterministic final reduction (fixed-order strided + LDS tree).
// ---------------------------------------------------------------------------
__global__ __launch_bounds__(256)
void reduce_kernel(const float* __restrict__ per,
                   const float* __restrict__ cnt,
                   float* __restrict__ out) {
  __shared__ float ssum[256];
  __shared__ float scnt[256];
  float s = 0.f, c = 0.f;
  for (int i = threadIdx.x; i < B_N; i += 256) { s += per[i]; c += cnt[i]; }
  ssum[threadIdx.x] = s;
  scnt[threadIdx.x] = c;
  __syncthreads();
  for (int off = 128; off > 0; off >>= 1) {
    if ((int)threadIdx.x < off) {
      ssum[threadIdx.x] += ssum[threadIdx.x + off];
      scnt[threadIdx.x] += scnt[threadIdx.x + off];
    }
    __syncthreads();
  }
  if (threadIdx.x == 0) {
    float cc = scnt[0];
    out[0] = (cc > 0.f) ? (ssum[0] / fmaxf(cc, 1.f)) : 0.f;
  }
}

// ---------------------------------------------------------------------------
extern "C" void kernel_launch(void* const* d_in, const int* in_sizes, int n_in,
                              void* d_out, int out_size, void* d_ws, size_t ws_size,
                              hipStream_t stream) {
  const float* emb    = (const float*)d_in[0];
  const int*   labels = (const int*)d_in[1];
  const int*   sbj    = (const int*)d_in[2];

  float* ws     = (float*)d_ws;
  float* sq     = ws;                      // [8192] f32
  int*   packed = (int*)(ws + 8192);       // [8192] i32
  int*   pidx   = (int*)(ws + 16384);      // [8192] i32
  int*   nidx   = (int*)(ws + 24576);      // [8192] i32
  float* per    = ws + 32768;              // [8192] f32
  float* cnt    = ws + 40960;              // [8192] f32

  precomp_kernel<<<B_N / 256, 256, 0, stream>>>(emb, labels, sbj, sq, packed);
  mine_kernel<<<B_N / 16, 256, 0, stream>>>(emb, sq, packed, pidx, nidx);
  triplet_kernel<<<B_N / 8, 256, 0, stream>>>(emb, pidx, nidx, per, cnt);
  reduce_kernel<<<1, 256, 0, stream>>>(per, cnt, (float*)d_out);
}